// SpikeFP32Linear_40973988004541
// MI455X (gfx1250) — compile-verified
//
#include <hip/hip_runtime.h>

typedef float v8f __attribute__((ext_vector_type(8)));
typedef float v4f __attribute__((ext_vector_type(4)));
typedef float v2f __attribute__((ext_vector_type(2)));

#define BATCH 256
#define INF   1024   // in_features
#define OUTF  1024   // out_features
#define NX    (BATCH * INF)    // 262144 packed x elements
#define NW    (OUTF * INF)     // 1048576 packed w elements

// ---------------------------------------------------------------------------
// Kernel 1: pack bit-pulses (32 floats, MSB first) -> one uint32 per element.
// Wave32 trick: lane j loads bit j of element e, v_cmp -> ballot_w32 gives the
// 32-bit mask with lane j at bit j; bit-reverse puts MSB-first order right
// (pulses[...,0] is bit 31). A lane-select accumulates the 32 packed scalars
// so the final store is one coalesced b32 per lane.
// ---------------------------------------------------------------------------
__global__ void spike_pack_kernel(const unsigned* __restrict__ xp,
                                  const unsigned* __restrict__ wp,
                                  unsigned* __restrict__ xf,
                                  unsigned* __restrict__ wf) {
    const int lane = threadIdx.x & 31;
    const int g    = blockIdx.x * (blockDim.x >> 5) + (threadIdx.x >> 5); // wave id
    const int base = g * 32;                                             // element base

    const unsigned* __restrict__ src;
    unsigned* __restrict__ dst;
    if (base < NX) {                       // uniform per wave (NX % 32 == 0)
        src = xp + (size_t)base * 32;
        dst = xf + base;
    } else {
        src = wp + (size_t)(base - NX) * 32;
        dst = wf + (base - NX);
    }

    unsigned acc = 0;
#pragma unroll
    for (int i = 0; i < 32; ++i) {
        unsigned bit = src[i * 32 + lane];                 // 0x0 or 0x3F800000
        unsigned m   = __builtin_amdgcn_ballot_w32(bit != 0u);
        m            = __builtin_bitreverse32(m);          // pulses[...,0] = MSB
        acc          = (lane == i) ? m : acc;              // v_cmp + v_cndmask
    }
    dst[lane] = acc;                                       // coalesced 128B store
}

// ---------------------------------------------------------------------------
// Kernel 2: fp32 GEMM via V_WMMA_F32_16X16X4_F32 + fused unpack epilogue.
// One wave = one 16x16 output tile (M = batch rows, N = out features).
// K loop in increasing order (matches the reference's left-to-right scan at
// 4-wide granularity). A fragment: 16x4 f32, lane = M (0-15) in each half,
// VGPR pair = consecutive k at k + 2*(lane>>4). B mirrors it with N = lane&15
// reading w_f rows (w is [OUT, IN] row major, B[k][n] = w[n][k]).
// ---------------------------------------------------------------------------
__global__ void spike_wmma_gemm_kernel(const float* __restrict__ xf,
                                       const float* __restrict__ wf,
                                       float* __restrict__ out) {
    const int lane = threadIdx.x & 31;
    const int wave = threadIdx.x >> 5;
    const int tile = blockIdx.x * (blockDim.x >> 5) + wave;   // 0..1023
    const int tm   = tile >> 6;            // 16 M-tiles
    const int tn   = tile & 63;            // 64 N-tiles
    const int m0   = tm * 16;
    const int n0   = tn * 16;

    const int half = lane >> 4;            // 0: K pair {0,1}, 1: K pair {2,3}
    const int l16  = lane & 15;

    const float* __restrict__ arow = xf + (size_t)(m0 + l16) * INF + 2 * half;
    const float* __restrict__ brow = wf + (size_t)(n0 + l16) * INF + 2 * half;

    v8f c = {};
#pragma unroll 8
    for (int k = 0; k < INF; k += 4) {
        v2f a = *(const v2f*)(arow + k);   // x_f[m, k+2h .. k+2h+1]
        v2f b = *(const v2f*)(brow + k);   // w_f[n, k+2h .. k+2h+1]
        // 8 args: (neg_a, A, neg_b, B, c_mod, C, reuse_a, reuse_b)
        c = __builtin_amdgcn_wmma_f32_16x16x4_f32(
                false, a, false, b, (short)0, c, false, false);
    }

    // Epilogue: unpack each fp32 result into 32 MSB-first pulse floats.
    // C/D layout: VGPR r, lanes 0-15 -> M=r, lanes 16-31 -> M=r+8; N = lane&15.
#pragma unroll
    for (int r = 0; r < 8; ++r) {
        const int row = m0 + r + 8 * half;          // batch index
        const int col = n0 + l16;                   // out-feature index
        const unsigned u = __float_as_uint(c[r]);
        float* __restrict__ o = out + ((size_t)row * OUTF + col) * 32;
#pragma unroll
        for (int q = 0; q < 8; ++q) {
            v4f f;
            f.x = (float)((u >> (31 - 4 * q)) & 1u);
            f.y = (float)((u >> (30 - 4 * q)) & 1u);
            f.z = (float)((u >> (29 - 4 * q)) & 1u);
            f.w = (float)((u >> (28 - 4 * q)) & 1u);
            *(v4f*)(o + 4 * q) = f;                 // 16B aligned
        }
    }
}

extern "C" void kernel_launch(void* const* d_in, const int* in_sizes, int n_in,
                              void* d_out, int out_size, void* d_ws, size_t ws_size,
                              hipStream_t stream) {
    const unsigned* x_pulses = (const unsigned*)d_in[0];   // [256,1024,32] f32 bits
    const unsigned* w_pulses = (const unsigned*)d_in[1];   // [1024,1024,32] f32 bits
    float* out = (float*)d_out;                            // [256,1024,32] f32

    unsigned* xf = (unsigned*)d_ws;        // 1 MB packed x
    unsigned* wf = xf + NX;                // 4 MB packed w

    // (NX + NW) / 32 waves = 40960 waves -> 5120 blocks of 8 waves (256 thr).
    spike_pack_kernel<<<(NX + NW) / 32 / 8, 256, 0, stream>>>(
        x_pulses, w_pulses, xf, wf);

    // 1024 tiles of 16x16, 8 waves per block -> 128 blocks.
    spike_wmma_gemm_kernel<<<1024 / 8, 256, 0, stream>>>(
        (const float*)xf, (const float*)wf, out);
}